// PhysicsInformedLoss_54503134986782
// MI455X (gfx1250) — compile-verified
//
#include <hip/hip_runtime.h>
#include <hip/hip_bf16.h>

#define BB 512
#define SS 8192

typedef float f4  __attribute__((ext_vector_type(4)));
typedef float v2f __attribute__((ext_vector_type(2)));
typedef float v8f __attribute__((ext_vector_type(8)));

// Wave32 full-sum reduction via V_WMMA_F32_16X16X4_F32.
// A: lane's partial in VGPR0 slot, 0 in VGPR1  (A[m][0]=x_m, A[m][2]=x_{m+16})
// B: all ones (layout-independent)
// => D[m][n] = x_m + x_{m+16} for every column n.
// D layout: lane n<16 holds D[0..7][n] in v0..7; lane n>=16 holds D[8..15][n-16].
// Per-lane sum of 8 D regs gives half-sum; xor-16 shuffle adds the other half.
__device__ __forceinline__ float wave_sum_wmma(float x) {
  v2f a; a[0] = x;    a[1] = 0.0f;
  v2f b; b[0] = 1.0f; b[1] = 1.0f;
  v8f c = {};
  v8f d = __builtin_amdgcn_wmma_f32_16x16x4_f32(false, a, false, b, (short)0, c,
                                                false, false);
  float s = d[0] + d[1] + d[2] + d[3] + d[4] + d[5] + d[6] + d[7];
  s += __shfl_xor(s, 16, 32);
  return s; // every lane holds the full 32-lane sum
}

// ---------------------------------------------------------------------------
// Kernel 1: one workgroup per batch row. Streams pred/target/strain once with
// non-temporal B128 loads and produces 8 per-row scalars in d_ws.
// ---------------------------------------------------------------------------
__global__ __launch_bounds__(256) void rowk(const float* __restrict__ pred,
                                            const float* __restrict__ target,
                                            const float* __restrict__ strain,
                                            const float* __restrict__ sscale,
                                            const float* __restrict__ smean,
                                            float* __restrict__ ws) {
  const int b   = blockIdx.x;
  const int tid = threadIdx.x;
  const float* prow = pred   + (size_t)b * SS;
  const float* trow = target + (size_t)b * SS;
  const float* erow = strain + (size_t)b * SS;
  const f4* p4 = (const f4*)prow;
  const f4* t4 = (const f4*)trow;
  const f4* e4 = (const f4*)erow;

  __shared__ float x0s[256];
  __shared__ float rsum[7][8];
  __shared__ float rmax[8];
  __shared__ float rmin[8];
  __shared__ int   ridx[8];

  float q[7];
#pragma unroll
  for (int i = 0; i < 7; ++i) q[i] = 0.0f;
  float pmax = -3.402823466e38f;
  float emin =  3.402823466e38f;
  int   eidx = 0;

  for (int it = 0; it < SS / 1024; ++it) {   // 8 iterations, uniform
    const int vi = it * 256 + tid;           // float4 index within row
    f4 p = __builtin_nontemporal_load(&p4[vi]);
    f4 t = __builtin_nontemporal_load(&t4[vi]);
    f4 e = __builtin_nontemporal_load(&e4[vi]);

    x0s[tid] = p[0];
    __syncthreads();
    float nxt = 0.0f;
    bool  hn  = true;
    if (tid < 255) {
      nxt = x0s[tid + 1];                    // neighbor's first element
    } else {
      const int ne = (vi + 1) * 4;           // first elem of next iter, thread 0
      if (ne < SS) nxt = prow[ne]; else hn = false;  // row end: no diff term
    }

    const int base = vi * 4;
#pragma unroll
    for (int k = 0; k < 4; ++k) {
      const float pk = p[k], tk = t[k], ek = e[k];
      const float d  = pk - tk;
      const float pw = 1.0f + 2.0f * __expf(ek * -50.0f); // 1 + LOW_W*exp(-e/LOW_SCALE)
      q[0] += d * d * pw;                                 // weighted MSE
      const float mk = (ek < 0.02f) ? 1.0f : 0.0f;        // elastic mask
      q[1] += mk * ek * pk;                               // sum strain*pred*mask
      q[2] += mk * ek;                                    // sum strain*mask
      q[3] += mk * ek * ek;                               // den
      q[4] += mk;                                         // cnt
      q[5] += pk;                                         // pred sum (for m)
      pmax = fmaxf(pmax, pk);
      const int idx = base + k;
      if (ek < emin) { emin = ek; eidx = idx; }           // first-occurrence argmin
      const float pn = (k < 3) ? p[k + 1] : nxt;
      if (k < 3 || hn) q[6] += fmaxf(pk - pn, 0.0f);      // mono (scale factored out)
    }
    __syncthreads();                                       // protect x0s reuse
  }

  // Cross-lane sums via WMMA, max/argmin via shuffles (wave32).
  float r[7];
#pragma unroll
  for (int i = 0; i < 7; ++i) r[i] = wave_sum_wmma(q[i]);
#pragma unroll
  for (int off = 16; off > 0; off >>= 1) {
    pmax = fmaxf(pmax, __shfl_xor(pmax, off, 32));
    const float oe = __shfl_xor(emin, off, 32);
    const int   oi = __shfl_xor(eidx, off, 32);
    if (oe < emin || (oe == emin && oi < eidx)) { emin = oe; eidx = oi; }
  }
  const int wid = tid >> 5, lane = tid & 31;
  if (lane == 0) {
#pragma unroll
    for (int i = 0; i < 7; ++i) rsum[i][wid] = r[i];
    rmax[wid] = pmax; rmin[wid] = emin; ridx[wid] = eidx;
  }
  __syncthreads();
  if (tid == 0) {
    float s[7];
#pragma unroll
    for (int i = 0; i < 7; ++i) {
      float acc = 0.0f;
      for (int w = 0; w < 8; ++w) acc += rsum[i][w];
      s[i] = acc;
    }
    float mx = rmax[0], mn = rmin[0];
    int   ix = ridx[0];
    for (int w = 1; w < 8; ++w) {
      mx = fmaxf(mx, rmax[w]);
      if (rmin[w] < mn || (rmin[w] == mn && ridx[w] < ix)) { mn = rmin[w]; ix = ridx[w]; }
    }
    const float scl = sscale[b], me = smean[b];   // scale > 0 by construction
    ws[0 * BB + b] = s[0];                        // sum d^2*pw (material w applied later)
    ws[1 * BB + b] = scl * s[1] + me * s[2];      // num = sum strain*s*mask
    ws[2 * BB + b] = s[3];                        // den
    ws[3 * BB + b] = s[4];                        // cnt
    ws[4 * BB + b] = scl * s[5] + (float)SS * me; // sum of s (for row mean m)
    ws[5 * BB + b] = scl * s[6];                  // mono sum
    ws[6 * BB + b] = scl * mx + me;               // max s
    ws[7 * BB + b] = scl * prow[ix] + me;         // s at argmin(strain)
  }
}

// ---------------------------------------------------------------------------
// Kernel 2: single workgroup. Per-row finalization + O(B^2) pairwise losses.
// ---------------------------------------------------------------------------
__global__ __launch_bounds__(256) void fink(const float* __restrict__ ws,
                                            const float* __restrict__ E_GPas,
                                            const float* __restrict__ rates,
                                            const float* __restrict__ temps,
                                            const int*   __restrict__ mats,
                                            float* __restrict__ out) {
  const int tid = threadIdx.x;
  __shared__ float m_sh[BB], t_sh[BB], r_sh[BB];
  __shared__ int   g_sh[BB];
  __shared__ float red[10][8];

  // material_weights = 71 / (counts*4), counts = {24,20,11,16}; UTS table
  const float matw[4] = {71.0f / 96.0f, 71.0f / 80.0f, 71.0f / 44.0f, 71.0f / 64.0f};
  const float uts[4]  = {470.0f, 400.0f, 310.0f, 570.0f};

  float a_mse = 0, a_rel = 0, a_val = 0, a_mono = 0, a_rng = 0, a_zero = 0;
  for (int b = tid; b < BB; b += 256) {
    const int g = mats[b];
    a_mse += matw[g] * ws[0 * BB + b];
    const float den = fmaxf(ws[2 * BB + b], 1e-8f);
    const float Ep  = ws[1 * BB + b] / den;
    const float Et  = E_GPas[b] * 1000.0f;
    const float rel = fabsf(Ep - Et) / fmaxf(Et, 1e-8f);
    const float valid = (ws[3 * BB + b] >= 3.0f) ? 1.0f : 0.0f;
    a_rel += rel * valid; a_val += valid;
    a_mono += ws[5 * BB + b];
    a_rng  += fmaxf(ws[6 * BB + b] - uts[g] * 1.2f, 0.0f);
    const float z = ws[7 * BB + b] / fmaxf(uts[g], 1e-6f);
    a_zero += z * z;
    m_sh[b] = ws[4 * BB + b] * (1.0f / (float)SS);
    t_sh[b] = temps[b]; r_sh[b] = rates[b]; g_sh[b] = g;
  }
  __syncthreads();

  // Pairwise masked means (upper triangle): 512*512/256 = 1024 uniform iters.
  float a_vt = 0, a_ct = 0, a_vr = 0, a_cr = 0;
  for (int p = tid; p < BB * BB; p += 256) {
    const int i = p >> 9, j = p & (BB - 1);
    if (j <= i || g_sh[i] != g_sh[j]) continue;
    const float Ti = t_sh[i], Tj = t_sh[j], Ri = r_sh[i], Rj = r_sh[j];
    const float mi = m_sh[i], mj = m_sh[j];
    if (fabsf(Ri - Rj) < 0.01f && Ti != Tj) {
      a_vt += fmaxf((Ti > Tj) ? (mi - mj) : (mj - mi), 0.0f); a_ct += 1.0f;
    }
    if (fabsf(Ti - Tj) < 5.0f && Ri != Rj) {
      a_vr += fmaxf((Ri > Rj) ? (mj - mi) : (mi - mj), 0.0f); a_cr += 1.0f;
    }
  }

  float v[10] = {a_mse, a_rel, a_val, a_mono, a_rng, a_zero, a_vt, a_ct, a_vr, a_cr};
#pragma unroll
  for (int i = 0; i < 10; ++i) v[i] = wave_sum_wmma(v[i]);
  const int wid = tid >> 5, lane = tid & 31;
  if (lane == 0) {
#pragma unroll
    for (int i = 0; i < 10; ++i) red[i][wid] = v[i];
  }
  __syncthreads();
  if (tid == 0) {
    float s[10];
#pragma unroll
    for (int i = 0; i < 10; ++i) {
      float acc = 0.0f;
      for (int w = 0; w < 8; ++w) acc += red[i][w];
      s[i] = acc;
    }
    const float mse_loss = s[0] / (float)((long long)BB * SS);
    const float elastic  = (s[2] > 0.0f) ? s[1] / fmaxf(s[2], 1.0f) : 0.0f;
    const float mono     = s[3] / (float)(BB * (SS - 1));
    const float rng      = s[4] / (float)BB;
    const float zero     = s[5] / (float)BB;
    const float templ    = (s[7] > 0.0f) ? s[6] / fmaxf(s[7], 1.0f) : 0.0f;
    const float ratel    = (s[9] > 0.0f) ? s[8] / fmaxf(s[9], 1.0f) : 0.0f;
    const float phys = elastic * 0.22f + mono * 0.18f + rng * 0.18f +
                       templ * 0.18f + ratel * 0.14f + zero * 0.1f;
    const float total = 0.7f * mse_loss + 0.3f * phys;
    out[0] = total;
    out[1] = mse_loss;
    out[2] = phys;
  }
}

extern "C" void kernel_launch(void* const* d_in, const int* in_sizes, int n_in,
                              void* d_out, int out_size, void* d_ws, size_t ws_size,
                              hipStream_t stream) {
  const float* pred   = (const float*)d_in[0];
  const float* target = (const float*)d_in[1];
  const float* strain = (const float*)d_in[2];
  const float* sscale = (const float*)d_in[3];
  const float* smean  = (const float*)d_in[4];
  const float* E      = (const float*)d_in[5];
  const float* rates  = (const float*)d_in[6];
  const float* temps  = (const float*)d_in[7];
  const int*   mats   = (const int*)d_in[8];
  float* out = (float*)d_out;
  float* ws  = (float*)d_ws;   // needs 8*512 floats = 16 KB

  rowk<<<BB, 256, 0, stream>>>(pred, target, strain, sscale, smean, ws);
  fink<<<1, 256, 0, stream>>>(ws, E, rates, temps, mats, out);
}